// GCNLayer_2705829396724
// MI455X (gfx1250) — compile-verified
//
#include <hip/hip_runtime.h>
#include <hip/hip_bf16.h>
#include <math.h>

#define IN_FEATS 128
#define OUT_FEATS 64
#define NEG_SLOPE 0.2f
#define SW_STRIDE 130   // 128 + 2 pad to avoid LDS bank conflicts

typedef float v2f __attribute__((ext_vector_type(2)));
typedef float v8f __attribute__((ext_vector_type(8)));

// ---------------------------------------------------------------------------
// Kernel 1: h = feat @ W^T using V_WMMA_F32_16X16X4_F32 (full fp32 precision).
// One wave -> 16 rows x 64 cols. Block = 256 threads = 8 waves = 128 rows.
// ---------------------------------------------------------------------------
__global__ void gat_gemm_wmma(const float* __restrict__ feat,
                              const float* __restrict__ W,
                              float* __restrict__ h, int n_nodes) {
  __shared__ float sW[OUT_FEATS * SW_STRIDE];  // 33,280 B of the 320 KB LDS
  // Cooperative stage of W [64 x 128] into padded LDS.
  for (int i = threadIdx.x; i < OUT_FEATS * IN_FEATS; i += blockDim.x) {
    int r = i >> 7, c = i & 127;
    sW[r * SW_STRIDE + c] = W[i];
  }
  __syncthreads();

  const int wave  = threadIdx.x >> 5;
  const int lane  = threadIdx.x & 31;
  const int mtile = blockIdx.x * 8 + wave;   // 16-row tile id
  const int row0  = mtile * 16;
  if (row0 >= n_nodes) return;               // wave-uniform exit

  const int mrow  = lane & 15;               // M (A) / N (B,D) position
  const int khalf = lane >> 4;               // K sub-pair select

  if (row0 + 16 <= n_nodes) {
    // Full tile: WMMA path (EXEC all ones).
    const float* arow = feat + (size_t)(row0 + mrow) * IN_FEATS;
    v8f acc0 = {}, acc1 = {}, acc2 = {}, acc3 = {};
    for (int k = 0; k < IN_FEATS; k += 4) {
      v2f a = *(const v2f*)(arow + k + 2 * khalf);
      v2f b0 = *(const v2f*)(sW + (0 * 16 + mrow) * SW_STRIDE + k + 2 * khalf);
      v2f b1 = *(const v2f*)(sW + (1 * 16 + mrow) * SW_STRIDE + k + 2 * khalf);
      v2f b2 = *(const v2f*)(sW + (2 * 16 + mrow) * SW_STRIDE + k + 2 * khalf);
      v2f b3 = *(const v2f*)(sW + (3 * 16 + mrow) * SW_STRIDE + k + 2 * khalf);
      acc0 = __builtin_amdgcn_wmma_f32_16x16x4_f32(false, a, false, b0, (short)0, acc0, false, false);
      acc1 = __builtin_amdgcn_wmma_f32_16x16x4_f32(false, a, false, b1, (short)0, acc1, false, false);
      acc2 = __builtin_amdgcn_wmma_f32_16x16x4_f32(false, a, false, b2, (short)0, acc2, false, false);
      acc3 = __builtin_amdgcn_wmma_f32_16x16x4_f32(false, a, false, b3, (short)0, acc3, false, false);
    }
    // D layout: VGPR r -> row (r + 8*khalf), col = (ntile*16 + mrow)
    float* hrow = h + (size_t)row0 * OUT_FEATS;
#pragma unroll
    for (int r = 0; r < 8; ++r) {
      size_t off = (size_t)(r + 8 * khalf) * OUT_FEATS + mrow;
      hrow[off + 0]  = acc0[r];
      hrow[off + 16] = acc1[r];
      hrow[off + 32] = acc2[r];
      hrow[off + 48] = acc3[r];
    }
  } else {
    // Scalar tail (not hit for n_nodes % 16 == 0, kept for safety).
    for (int r = row0 + lane; r < n_nodes; r += 32) {
      for (int n = 0; n < OUT_FEATS; ++n) {
        float acc = 0.f;
        for (int k = 0; k < IN_FEATS; ++k)
          acc += feat[(size_t)r * IN_FEATS + k] * sW[n * SW_STRIDE + k];
        h[(size_t)r * OUT_FEATS + n] = acc;
      }
    }
  }
}

// ---------------------------------------------------------------------------
// Kernel 2: init h_agg = 0, m = -inf, s = 0 (buffers are poisoned each call).
// ---------------------------------------------------------------------------
__global__ void gat_init(float* __restrict__ h_agg, float* __restrict__ m,
                         float* __restrict__ s, int n_nodes) {
  int i = blockIdx.x * blockDim.x + threadIdx.x;
  int tot = n_nodes * OUT_FEATS;
  if (i < tot) h_agg[i] = 0.0f;
  if (i < n_nodes) { m[i] = -INFINITY; s[i] = 0.0f; }
}

// ---------------------------------------------------------------------------
// Kernel 3: h_agg[dst[e]] += h[src[e]]  (64 lanes per edge, f32 atomics in L2)
// ---------------------------------------------------------------------------
__global__ void gat_scatter(const float* __restrict__ h, const int* __restrict__ src,
                            const int* __restrict__ dst, float* __restrict__ h_agg,
                            int n_edges) {
  unsigned int tid = blockIdx.x * blockDim.x + threadIdx.x;
  unsigned int e = tid >> 6;
  unsigned int j = tid & 63u;
  if (e >= (unsigned int)n_edges) return;
  int su = src[e], dv = dst[e];
  atomicAdd(&h_agg[(size_t)dv * OUT_FEATS + j], h[(size_t)su * OUT_FEATS + j]);
}

// ---------------------------------------------------------------------------
// Kernel 4: per-edge score, one wave32 per edge, shuffle reduction.
// ---------------------------------------------------------------------------
__global__ void gat_score(const float* __restrict__ h_agg, const int* __restrict__ src,
                          const int* __restrict__ dst, float* __restrict__ escore,
                          int n_edges) {
  int wid  = (blockIdx.x * blockDim.x + threadIdx.x) >> 5;
  int lane = threadIdx.x & 31;
  if (wid >= n_edges) return;          // wave-uniform
  int su = src[wid], dv = dst[wid];
  const float* hs = h_agg + (size_t)su * OUT_FEATS;
  const float* hd = h_agg + (size_t)dv * OUT_FEATS;
  float p = hs[lane] * tanhf(hd[lane]) + hs[lane + 32] * tanhf(hd[lane + 32]);
#pragma unroll
  for (int off = 16; off > 0; off >>= 1)
    p += __shfl_down(p, off, 32);
  if (lane == 0) {
    escore[wid] = p > 0.0f ? p : NEG_SLOPE * p;   // leaky_relu
  }
}

// ---------------------------------------------------------------------------
// Kernel 5: m[dst] = segment_max(e) via sign-aware int atomic max/min trick.
// ---------------------------------------------------------------------------
__global__ void gat_segmax(const float* __restrict__ escore, const int* __restrict__ dst,
                           float* __restrict__ m, int n_edges) {
  int i = blockIdx.x * blockDim.x + threadIdx.x;
  if (i >= n_edges) return;
  float v = escore[i];
  float* addr = &m[dst[i]];
  if (v >= 0.0f) atomicMax((int*)addr, __float_as_int(v));
  else           atomicMin((unsigned int*)addr, __float_as_uint(v));
}

// ---------------------------------------------------------------------------
// Kernel 6: ex = exp(e - m[dst]); s[dst] += ex
// ---------------------------------------------------------------------------
__global__ void gat_expsum(const float* __restrict__ escore, const int* __restrict__ dst,
                           const float* __restrict__ m, float* __restrict__ ex,
                           float* __restrict__ ssum, int n_edges) {
  int i = blockIdx.x * blockDim.x + threadIdx.x;
  if (i >= n_edges) return;
  int d = dst[i];
  float x = expf(escore[i] - m[d]);
  ex[i] = x;
  atomicAdd(&ssum[d], x);
}

// ---------------------------------------------------------------------------
// Kernel 7: e_soft = ex / s[dst]
// ---------------------------------------------------------------------------
__global__ void gat_norm(const float* __restrict__ ex, const int* __restrict__ dst,
                         const float* __restrict__ ssum, float* __restrict__ out,
                         int n_edges) {
  int i = blockIdx.x * blockDim.x + threadIdx.x;
  if (i >= n_edges) return;
  out[i] = ex[i] / ssum[dst[i]];
}

// ---------------------------------------------------------------------------
extern "C" void kernel_launch(void* const* d_in, const int* in_sizes, int n_in,
                              void* d_out, int out_size, void* d_ws, size_t ws_size,
                              hipStream_t stream) {
  const float* feat = (const float*)d_in[0];
  const float* W    = (const float*)d_in[1];
  const int*   src  = (const int*)d_in[2];
  const int*   dst  = (const int*)d_in[3];

  const int n_nodes = in_sizes[0] / IN_FEATS;  // 100000
  const int n_edges = in_sizes[2];             // 1600000

  // Output layout: [h_agg : n_nodes*64][e_soft : n_edges]
  float* h_agg  = (float*)d_out;
  float* e_soft = (float*)d_out + (size_t)n_nodes * OUT_FEATS;

  // Workspace layout
  float* ws = (float*)d_ws;
  float* h  = ws;                                    // n_nodes*64
  float* es = h  + (size_t)n_nodes * OUT_FEATS;      // n_edges
  float* ex = es + n_edges;                          // n_edges
  float* m  = ex + n_edges;                          // n_nodes
  float* s  = m  + n_nodes;                          // n_nodes

  // 1) GEMM (WMMA f32): 8 waves/block -> 128 rows/block
  gat_gemm_wmma<<<(n_nodes + 127) / 128, 256, 0, stream>>>(feat, W, h, n_nodes);

  // 2) init h_agg / m / s
  {
    unsigned int tot = (unsigned int)n_nodes * OUT_FEATS;
    gat_init<<<(tot + 255) / 256, 256, 0, stream>>>(h_agg, m, s, n_nodes);
  }

  // 3) scatter-add (64 threads per edge)
  {
    unsigned int tot = (unsigned int)n_edges * 64u;
    gat_scatter<<<(tot + 255) / 256, 256, 0, stream>>>(h, src, dst, h_agg, n_edges);
  }

  // 4) edge scores (1 wave per edge)
  {
    unsigned int tot = (unsigned int)n_edges * 32u;
    gat_score<<<(tot + 255) / 256, 256, 0, stream>>>(h_agg, src, dst, es, n_edges);
  }

  // 5..7) edge softmax
  gat_segmax<<<(n_edges + 255) / 256, 256, 0, stream>>>(es, dst, m, n_edges);
  gat_expsum<<<(n_edges + 255) / 256, 256, 0, stream>>>(es, dst, m, ex, s, n_edges);
  gat_norm  <<<(n_edges + 255) / 256, 256, 0, stream>>>(ex, dst, s, e_soft, n_edges);
}